// Adder2D_43258910605605
// MI455X (gfx1250) — compile-verified
//
#include <hip/hip_runtime.h>

// ---------------------------------------------------------------------------
// AdderNet 2D "conv": out[n,co,ho,wo] = -sum_{ci,kh,kw} |x_pad - w|
// fp32, VALU-bound (sum of |a-b| is not bilinear -> WMMA inapplicable).
// CDNA5 paths: BOTH x and w tiles streamed by the Tensor Data Mover
// (tensor_load_to_lds, TENSORcnt), double-buffered in LDS and software-
// pipelined against the VALU hot loop; s_wait_tensorcnt for sync.
// ---------------------------------------------------------------------------

typedef unsigned int u32x4 __attribute__((ext_vector_type(4)));
typedef int          i32x4 __attribute__((ext_vector_type(4)));
typedef int          i32x8 __attribute__((ext_vector_type(8)));

#define NN 16
#define CC_IN 64
#define HH 56
#define WW 56
#define CO_ALL 64
#define KK 3
#define TH 4             // output rows per block
#define CI_CHUNK 16      // ci per LDS chunk
#define CO_BLK 32        // co per block (grid.z = 2)
#define XROWS (TH + 2)   // 6 input rows per tile
#define NTHREADS 224     // 7 waves; 56 spatial slots x 4 co sublanes

#define XTILE (CI_CHUNK * XROWS * WW)            // 5376 floats
#define WTILE (CO_BLK * CI_CHUNK * KK * KK)      // 4608 floats

// Build a TDM D# (2-D or 3-D tile of 4-byte elements) and issue the load.
__device__ __forceinline__ void tdm_load(unsigned lds_byte_off, const void* gptr,
                                         unsigned tensor_d0, unsigned tensor_d1,
                                         unsigned tensor_d2,
                                         unsigned tile_d0, unsigned tile_d1,
                                         unsigned tile_d2,
                                         unsigned long long stride0,
                                         unsigned long long stride1)
{
    unsigned long long ga = (unsigned long long)gptr;
    u32x4 g0;
    g0[0] = 1u;                                        // count=1, no gather
    g0[1] = lds_byte_off;                              // lds_addr (bytes)
    g0[2] = (unsigned)ga;                              // global_addr[31:0]
    g0[3] = (unsigned)((ga >> 32) & 0x01FFFFFFull) | (2u << 30);  // type=2
    i32x8 g1;
    g1[0] = (int)(2u << 16);                           // data_size = 4 bytes
    g1[1] = (int)(tensor_d0 << 16);                    // tensor_dim0 lo16
    g1[2] = (int)((tensor_d0 >> 16) | (tensor_d1 << 16));
    g1[3] = (int)((tensor_d1 >> 16) | (tile_d0 << 16));
    g1[4] = (int)(tile_d1 | (tile_d2 << 16));
    g1[5] = (int)(unsigned)(stride0 & 0xFFFFFFFFull);
    g1[6] = (int)((unsigned)((stride0 >> 32) & 0xFFFFull) |
                  ((unsigned)(stride1 & 0xFFFFull) << 16));
    g1[7] = (int)(unsigned)((stride1 >> 16) & 0xFFFFFFFFull);
    i32x4 g2;
    g2[0] = (int)tensor_d2; g2[1] = 0; g2[2] = 0; g2[3] = 0;
    i32x4 g3 = {0, 0, 0, 0};
#if defined(__clang_major__) && __clang_major__ >= 23
    i32x8 z8 = {0, 0, 0, 0, 0, 0, 0, 0};
    __builtin_amdgcn_tensor_load_to_lds(g0, g1, g2, g3, z8, 0);
#else
    __builtin_amdgcn_tensor_load_to_lds(g0, g1, g2, g3, 0);
#endif
}

__global__ __launch_bounds__(NTHREADS)
void adder2d_kernel(const float* __restrict__ x,
                    const float* __restrict__ w,
                    float* __restrict__ out)
{
    __shared__ float xlds[2][XTILE];   // 43,008 B (double-buffered)
    __shared__ float wlds[2][WTILE];   // 36,864 B (double-buffered)

    const int tid   = threadIdx.x;
    const int cosub = tid & 3;          // 4 co sub-lanes
    const int slot  = tid >> 2;         // 0..55 spatial slot
    const int ho_r  = slot / 14;        // 0..3
    const int wo0   = (slot % 14) * 4;  // 4 consecutive wo per thread
    const int ho0   = blockIdx.x * TH;
    const int n     = blockIdx.y;
    const int cob   = blockIdx.z * CO_BLK;

    // Clamped tile start row: tile is ALWAYS 6 fully in-bounds rows.
    int h_start = ho0 - 1; if (h_start < 0) h_start = 0; if (h_start > HH - XROWS) h_start = HH - XROWS;

    // Per-thread halo masks/indices (loop-invariant for the whole kernel).
    int c_idx[6], c_ok[6];
#pragma unroll
    for (int j = 0; j < 6; ++j) {
        int c = wo0 - 1 + j;
        c_ok[j]  = ((unsigned)c < (unsigned)WW) ? 1 : 0;
        c_idx[j] = c < 0 ? 0 : (c > WW - 1 ? WW - 1 : c);
    }
    int r_idx[3], r_ok[3];
#pragma unroll
    for (int kh = 0; kh < KK; ++kh) {
        int g  = ho0 + ho_r + kh - 1;          // global input row needed
        int rr = g - h_start;                  // row within the 6-row tile
        r_ok[kh]  = ((unsigned)g < (unsigned)HH) ? 1 : 0;
        r_idx[kh] = rr < 0 ? 0 : (rr > XROWS - 1 ? XROWS - 1 : rr);
    }

    float acc[8][4];
#pragma unroll
    for (int cr = 0; cr < 8; ++cr)
#pragma unroll
        for (int s = 0; s < 4; ++s) acc[cr][s] = 0.0f;

    const float* xn = x + (size_t)n * CC_IN * HH * WW;
    const unsigned lds_x0 = (unsigned)(unsigned long long)(void*)&xlds[0][0];
    const unsigned lds_x1 = (unsigned)(unsigned long long)(void*)&xlds[1][0];
    const unsigned lds_w0 = (unsigned)(unsigned long long)(void*)&wlds[0][0];
    const unsigned lds_w1 = (unsigned)(unsigned long long)(void*)&wlds[1][0];

    // ---- preload chunk 0 into buffer 0 (wave 0 drives the TDM) ------------
    if (tid < 32) {
        tdm_load(lds_w0, w + (size_t)cob * (CC_IN * KK * KK),
                 576, 64, 0, CI_CHUNK * KK * KK, CO_BLK, 0, 576, 0);
        tdm_load(lds_x0, xn + (size_t)h_start * WW,
                 WW, HH, CC_IN, WW, XROWS, CI_CHUNK, WW, (unsigned long long)HH * WW);
    }

#pragma unroll 1
    for (int cc = 0; cc < CC_IN / CI_CHUNK; ++cc) {
        if (tid < 32) __builtin_amdgcn_s_wait_tensorcnt(0);
        __syncthreads();   // chunk cc resident; all waves done reading cc-1's buffer

        // Prefetch chunk cc+1 into the other buffer while everyone computes.
        if (tid < 32 && cc + 1 < CC_IN / CI_CHUNK) {
            const int ci1 = (cc + 1) * CI_CHUNK;
            const unsigned lw = ((cc + 1) & 1) ? lds_w1 : lds_w0;
            const unsigned lx = ((cc + 1) & 1) ? lds_x1 : lds_x0;
            tdm_load(lw, w + (size_t)cob * (CC_IN * KK * KK) + (size_t)ci1 * (KK * KK),
                     576, 64, 0, CI_CHUNK * KK * KK, CO_BLK, 0, 576, 0);
            tdm_load(lx, xn + (size_t)ci1 * (HH * WW) + (size_t)h_start * WW,
                     WW, HH, CC_IN, WW, XROWS, CI_CHUNK, WW, (unsigned long long)HH * WW);
        }

        // ---- hot loop on buffer cc&1: pure ds_load + v_sub/v_add(|.|) -----
        const float* xbuf = &xlds[cc & 1][0];
        const float* wco  = &wlds[cc & 1][cosub * (CI_CHUNK * KK * KK)];
#pragma unroll 1
        for (int ci_r = 0; ci_r < CI_CHUNK; ++ci_r) {
            const float* xci = xbuf + ci_r * (XROWS * WW);
            const float* wb  = wco + ci_r * (KK * KK);
#pragma unroll
            for (int kh = 0; kh < KK; ++kh) {
                const float* xr = xci + r_idx[kh] * WW;
                float xv[6];
#pragma unroll
                for (int j = 0; j < 6; ++j) {
                    float v = xr[c_idx[j]];
                    xv[j] = (r_ok[kh] & c_ok[j]) ? v : 0.0f;   // zero-pad halo
                }
#pragma unroll
                for (int kw = 0; kw < KK; ++kw) {
#pragma unroll
                    for (int cr = 0; cr < 8; ++cr) {
                        // co = cob + cr*4 + cosub; cosub strides 144 dwords
                        // -> banks 16 apart across the 4 sublanes (conflict-free)
                        float wv = wb[cr * (4 * CI_CHUNK * KK * KK) + kh * KK + kw];
#pragma unroll
                        for (int s = 0; s < 4; ++s)
                            acc[cr][s] += __builtin_fabsf(xv[s + kw] - wv);
                    }
                }
            }
        }
    }

    // ---- write out: one float4 per co ------------------------------------
    float* outn = out + (size_t)n * CO_ALL * HH * WW;
    const int orow = (ho0 + ho_r) * WW + wo0;
#pragma unroll
    for (int cr = 0; cr < 8; ++cr) {
        int co = cob + cr * 4 + cosub;
        float4 v = make_float4(-acc[cr][0], -acc[cr][1], -acc[cr][2], -acc[cr][3]);
        *(float4*)&outn[(size_t)co * (HH * WW) + orow] = v;
    }
}

extern "C" void kernel_launch(void* const* d_in, const int* in_sizes, int n_in,
                              void* d_out, int out_size, void* d_ws, size_t ws_size,
                              hipStream_t stream) {
    const float* x = (const float*)d_in[0];
    const float* w = (const float*)d_in[1];
    float* out = (float*)d_out;
    dim3 grid(HH / TH, NN, CO_ALL / CO_BLK);   // (14, 16, 2) = 448 blocks
    adder2d_kernel<<<grid, NTHREADS, 0, stream>>>(x, w, out);
}